// GNNClassifier_73985106641261
// MI455X (gfx1250) — compile-verified
//
#include <hip/hip_runtime.h>
#include <hip/hip_bf16.h>

typedef __attribute__((ext_vector_type(2))) float v2f;
typedef __attribute__((ext_vector_type(8))) float v8f;

#define IN_DIM   128
#define HID_DIM  128
#define OUT_DIM  10
#define N_GRAPHS 512
#define LDS_STRIDE 132   // 128 + 4 pad: conflict-free ds_load_b64 across all 32 lanes

// ---------------------------------------------------------------------------
// degree / norm
// ---------------------------------------------------------------------------
__global__ void deg_init_kernel(float* __restrict__ deg, int n) {
    int i = blockIdx.x * blockDim.x + threadIdx.x;
    if (i < n) deg[i] = 1.0f;   // self-loop contributes 1
}

__global__ void deg_accum_kernel(const int* __restrict__ dst, float* __restrict__ deg, int ne) {
    int e = blockIdx.x * blockDim.x + threadIdx.x;
    if (e < ne) atomicAdd(&deg[dst[e]], 1.0f);
}

__global__ void dinv_kernel(const float* __restrict__ deg, float* __restrict__ dinv, int n) {
    int i = blockIdx.x * blockDim.x + threadIdx.x;
    if (i < n) dinv[i] = rsqrtf(fmaxf(deg[i], 1.0f));
}

// ---------------------------------------------------------------------------
// H = X @ W   (X: [n,128], W: [128,128] row-major), V_WMMA_F32_16X16X4_F32.
// Block = 256 threads (8 waves): 16 rows x 128 cols. W transposed into LDS
// (B pairs contiguous -> single ds_load_b64), X tile staged in LDS (shared A).
// ---------------------------------------------------------------------------
__global__ __launch_bounds__(256) void gemm_x128_wmma(
        const float* __restrict__ X, const float* __restrict__ W,
        float* __restrict__ H, int nrows) {
    __shared__ float Wt[128 * LDS_STRIDE];  // Wt[col][k] = W[k][col]
    __shared__ float Xs[16 * LDS_STRIDE];   // X tile rows

    const int tid = threadIdx.x;
    const int row_base = blockIdx.x << 4;

    // stage W transposed: coalesced float4 reads, scattered scalar writes
    {
        const float4* Wv = (const float4*)W;
        for (int i = tid; i < (128 * 128) / 4; i += 256) {
            int r  = i >> 5;           // K row of W: 0..127
            int c4 = (i & 31) << 2;    // col: 0,4,...,124
            float4 v = Wv[i];
            Wt[(c4 + 0) * LDS_STRIDE + r] = v.x;
            Wt[(c4 + 1) * LDS_STRIDE + r] = v.y;
            Wt[(c4 + 2) * LDS_STRIDE + r] = v.z;
            Wt[(c4 + 3) * LDS_STRIDE + r] = v.w;
        }
    }
    // stage X tile (16 rows x 128): 2 float4 per thread
    {
        for (int i = tid; i < 16 * 32; i += 256) {
            int r  = i >> 5;
            int c4 = i & 31;
            int gr = row_base + r;
            if (gr >= nrows) gr = nrows - 1;   // clamp; stores guarded below
            float4 v = ((const float4*)(X + (size_t)gr * IN_DIM))[c4];
            *(float4*)(&Xs[r * LDS_STRIDE + (c4 << 2)]) = v;
        }
    }
    __syncthreads();

    const int wave = tid >> 5;             // 0..7 -> column tile
    const int lane = tid & 31;
    const int half = lane >> 4;            // 0: K {0,1}, 1: K {2,3} / M+8
    const int l16  = lane & 15;
    const int col_base = wave << 4;

    const float* xa = Xs + l16 * LDS_STRIDE + 2 * half;                 // A: row l16
    const float* wb = Wt + (col_base + l16) * LDS_STRIDE + 2 * half;    // B: col l16

    v8f acc = {};
#pragma unroll
    for (int k = 0; k < 128; k += 4) {
        v2f a = *(const v2f*)(xa + k);   // ds_load_b64, conflict-free
        v2f b = *(const v2f*)(wb + k);   // ds_load_b64, conflict-free
        acc = __builtin_amdgcn_wmma_f32_16x16x4_f32(
                /*neg_a=*/false, a, /*neg_b=*/false, b,
                /*c_mod=*/(short)0, acc, /*reuse_a=*/false, /*reuse_b=*/false);
    }

    // D 16x16: VGPR j -> M = j + 8*half, N = l16
#pragma unroll
    for (int j = 0; j < 8; ++j) {
        int r = row_base + 8 * half + j;
        if (r < nrows) H[(size_t)r * HID_DIM + col_base + l16] = acc[j];
    }
}

// ---------------------------------------------------------------------------
// Out[i][f] = H[i][f] * dinv[i]^2 + b[f]    (self-loop message + bias)
// ---------------------------------------------------------------------------
__global__ void self_bias_kernel(const float* __restrict__ H, const float* __restrict__ dinv,
                                 const float* __restrict__ b, float* __restrict__ Out,
                                 int n) {
    long long i = (long long)blockIdx.x * blockDim.x + threadIdx.x;
    if (i >= (long long)n * HID_DIM) return;
    int node = (int)(i >> 7);
    int f = (int)(i & 127);
    float di = dinv[node];
    Out[i] = H[i] * di * di + b[f];
}

// ---------------------------------------------------------------------------
// edge scatter: Out[dst] += H[src] * dinv[src]*dinv[dst]
// one wave per edge, each lane handles 4 features (float4)
// ---------------------------------------------------------------------------
__global__ __launch_bounds__(256) void edge_scatter_kernel(
        const float* __restrict__ H, const int* __restrict__ src,
        const int* __restrict__ dst, const float* __restrict__ dinv,
        float* __restrict__ Out, int ne) {
    int e = blockIdx.x * (blockDim.x >> 5) + (threadIdx.x >> 5);
    if (e >= ne) return;
    int lane = threadIdx.x & 31;
    int s = src[e], d = dst[e];
    float w = dinv[s] * dinv[d];
    float4 v = *((const float4*)(H + (size_t)s * HID_DIM) + lane);
    float* o = Out + (size_t)d * HID_DIM + lane * 4;
    atomicAdd(o + 0, v.x * w);
    atomicAdd(o + 1, v.y * w);
    atomicAdd(o + 2, v.z * w);
    atomicAdd(o + 3, v.w * w);
}

__global__ void relu_kernel(float* __restrict__ A, long long total) {
    long long i = (long long)blockIdx.x * blockDim.x + threadIdx.x;
    if (i < total) A[i] = fmaxf(A[i], 0.0f);
}

// ---------------------------------------------------------------------------
// pooling
// ---------------------------------------------------------------------------
__global__ void pool_zero_kernel(float* __restrict__ pool, float* __restrict__ cnt) {
    int i = blockIdx.x * blockDim.x + threadIdx.x;
    if (i < N_GRAPHS * HID_DIM) pool[i] = 0.0f;
    if (i < N_GRAPHS) cnt[i] = 0.0f;
}

__global__ __launch_bounds__(256) void pool_accum_kernel(
        const float* __restrict__ H, const int* __restrict__ batch,
        float* __restrict__ pool, float* __restrict__ cnt, int n) {
    int node = blockIdx.x * (blockDim.x >> 5) + (threadIdx.x >> 5);
    if (node >= n) return;
    int lane = threadIdx.x & 31;
    int g = batch[node];
    float4 v = *((const float4*)(H + (size_t)node * HID_DIM) + lane);
    float* p = pool + (size_t)g * HID_DIM + lane * 4;
    atomicAdd(p + 0, v.x);
    atomicAdd(p + 1, v.y);
    atomicAdd(p + 2, v.z);
    atomicAdd(p + 3, v.w);
    if (lane == 0) atomicAdd(&cnt[g], 1.0f);
}

// ---------------------------------------------------------------------------
// head: out[g][o] = (pool[g]/max(cnt,1)) . Wp[:,o] + bp[o]
// ---------------------------------------------------------------------------
__global__ void head_kernel(const float* __restrict__ pool, const float* __restrict__ cnt,
                            const float* __restrict__ Wp, const float* __restrict__ bp,
                            float* __restrict__ out) {
    int idx = blockIdx.x * blockDim.x + threadIdx.x;
    if (idx >= N_GRAPHS * OUT_DIM) return;
    int g = idx / OUT_DIM;
    int o = idx - g * OUT_DIM;
    float inv = 1.0f / fmaxf(cnt[g], 1.0f);
    const float* p = pool + (size_t)g * HID_DIM;
    float s = 0.0f;
#pragma unroll 8
    for (int f = 0; f < HID_DIM; ++f) s = fmaf(p[f] * inv, Wp[f * OUT_DIM + o], s);
    out[idx] = s + bp[o];
}

// ---------------------------------------------------------------------------
// launch
// ---------------------------------------------------------------------------
extern "C" void kernel_launch(void* const* d_in, const int* in_sizes, int n_in,
                              void* d_out, int out_size, void* d_ws, size_t ws_size,
                              hipStream_t stream) {
    const float* x     = (const float*)d_in[0];
    const int*   eidx  = (const int*)d_in[1];   // [2, E] flat: src then dst
    const int*   batch = (const int*)d_in[2];
    const float* W1    = (const float*)d_in[3];
    const float* b1    = (const float*)d_in[4];
    const float* W2    = (const float*)d_in[5];
    const float* b2    = (const float*)d_in[6];
    const float* Wp    = (const float*)d_in[7];
    const float* bp    = (const float*)d_in[8];
    float* out = (float*)d_out;

    const int n  = in_sizes[0] / IN_DIM;     // 50000
    const int ne = in_sizes[1] / 2;          // 800000
    const int* esrc = eidx;
    const int* edst = eidx + ne;

    // workspace carve-up (floats)
    float* deg  = (float*)d_ws;                        // n
    float* dinv = deg + n;                             // n
    float* pool = dinv + n;                            // 512*128
    float* cnt  = pool + (size_t)N_GRAPHS * HID_DIM;   // 512
    float* bufH = cnt + N_GRAPHS;                      // n*128 (GEMM result of each layer)
    float* bufO = bufH + (size_t)n * HID_DIM;          // n*128 (conv output of each layer)

    const long long nfeat = (long long)n * HID_DIM;
    const int tb = 256;
    const int gN     = (n + tb - 1) / tb;
    const int gE     = (ne + tb - 1) / tb;
    const int gFeat  = (int)((nfeat + tb - 1) / tb);
    const int gTiles = (n + 15) / 16;                  // 3125
    const int gWaveE = (ne + 7) / 8;                   // 1 wave/edge, 8 waves/block
    const int gWaveN = (n + 7) / 8;

    // ---- normalization (shared by both conv layers) ----
    deg_init_kernel<<<gN, tb, 0, stream>>>(deg, n);
    deg_accum_kernel<<<gE, tb, 0, stream>>>(edst, deg, ne);
    dinv_kernel<<<gN, tb, 0, stream>>>(deg, dinv, n);

    // ---- layer 1: relu(conv(x, W1, b1)) ----
    gemm_x128_wmma<<<gTiles, tb, 0, stream>>>(x, W1, bufH, n);
    self_bias_kernel<<<gFeat, tb, 0, stream>>>(bufH, dinv, b1, bufO, n);
    edge_scatter_kernel<<<gWaveE, tb, 0, stream>>>(bufH, esrc, edst, dinv, bufO, ne);
    relu_kernel<<<gFeat, tb, 0, stream>>>(bufO, nfeat);

    // ---- layer 2: conv(h, W2, b2) ----
    gemm_x128_wmma<<<gTiles, tb, 0, stream>>>(bufO, W2, bufH, n);
    self_bias_kernel<<<gFeat, tb, 0, stream>>>(bufH, dinv, b2, bufO, n);   // bufO free after gemm2
    edge_scatter_kernel<<<gWaveE, tb, 0, stream>>>(bufH, esrc, edst, dinv, bufO, ne);

    // ---- global mean pool + head ----
    pool_zero_kernel<<<(N_GRAPHS * HID_DIM + tb - 1) / tb, tb, 0, stream>>>(pool, cnt);
    pool_accum_kernel<<<gWaveN, tb, 0, stream>>>(bufO, batch, pool, cnt, n);
    head_kernel<<<(N_GRAPHS * OUT_DIM + tb - 1) / tb, tb, 0, stream>>>(pool, cnt, Wp, bp, out);
}